// PKM_87703232185060
// MI455X (gfx1250) — compile-verified
//
#include <hip/hip_runtime.h>
#include <math.h>

// PKM (product-key memory) for MI455X / gfx1250.
// (A) q = x@Wq^T+bq   : fp32 WMMA, B-tile staged to LDS by the Tensor Data Mover
// (B) scores = q@keys^T: same structure, 16 batched GEMMs
// (C) dual top-16 -> 16x16 product top-16 -> softmax
// (D) gather + weighted sum (HBM-bound, ~1.07 GB => ~46 us floor @ 23.3 TB/s)
// Workspace requirement: ~98 MiB.

typedef float v2f __attribute__((ext_vector_type(2)));
typedef float v8f __attribute__((ext_vector_type(8)));
typedef unsigned int v4u __attribute__((ext_vector_type(4)));
typedef int v8i __attribute__((ext_vector_type(8)));
typedef int v4i __attribute__((ext_vector_type(4)));

#define DIMX   1024
#define NKEYS  512
#define HEADS  8
#define KNN    16
#define NTOK   2048
#define QCOLS  4096   // HEADS * K_DIM
#define KHALF  256
#define KC     64     // K-chunk staged per TDM transfer
#define LDSP   66     // LDS row stride in floats (64 data + 2 TDM pad dwords)
#define TILE_LDS_BYTES (64 * LDSP * 4)   // 16896 B dynamic LDS, tile at offset 0

// ---------------------------------------------------------------------------
// TDM: load a [64 rows x 64 floats] tile of a row-major fp32 matrix
// (row stride = stride0 floats) into LDS offset 0, padding each 64-DWORD row
// with 2 DWORDs (row stride LDSP floats in LDS -> <=2-way bank conflicts on
// the ds_load_b64 fragment reads). D# per CDNA5 ISA.
// ---------------------------------------------------------------------------
__device__ __forceinline__ void tdm_load_tile_f32(const float* gptr,
                                                  unsigned tensor_d0,
                                                  unsigned tensor_d1,
                                                  unsigned stride0)
{
    const unsigned long long ga = (unsigned long long)(size_t)gptr;
    v4u g0;
    g0.x = 1u;                                        // [1:0] count=1 (user D#)
    g0.y = 0u;                                        // [63:32] lds_addr = 0
    g0.z = (unsigned)(ga & 0xFFFFFFFFull);            // [95:64] global_addr lo
    g0.w = (unsigned)((ga >> 32) & 0x1FFFFFFull)      // [120:96] global_addr hi
         | 0x80000000u;                               // [127:126] type=2 (image)
    v8i g1;
    g1[0] = (int)((2u << 16)        // [17:16] data_size = 4 bytes
                | (1u << 20)        // [20]    pad_enable
                | (5u << 22)        // [24:22] pad_interval: 2^(5+1)=64 DWORDs
                | (1u << 25));      // [31:25] pad_amount: 1+1 = 2 DWORDs
    g1[1] = (int)((tensor_d0 & 0xFFFFu) << 16);                    // dim0 lo16
    g1[2] = (int)((tensor_d0 >> 16) | ((tensor_d1 & 0xFFFFu) << 16));
    g1[3] = (int)((tensor_d1 >> 16) | ((unsigned)KC << 16));       // tile_dim0
    g1[4] = (int)KC;                                               // tile_dim1 (=64 rows), tile_dim2=0
    g1[5] = (int)stride0;                                          // dim0_stride lo32
    g1[6] = 0;
    g1[7] = 0;
    const v4i z4 = {0, 0, 0, 0};                                   // 2-D: groups 2/3 unused
    const v8i z8 = {0, 0, 0, 0, 0, 0, 0, 0};
    __builtin_amdgcn_tensor_load_to_lds(g0, g1, z4, z4, z8, 0);
}

// ---------------------------------------------------------------------------
// Read the four B fragments of one k-step straight from LDS with real
// ds_load_b64 ops (clang's AA cannot see the TDM's LDS writes, so C-level
// loads get folded to undef / or, if volatile, lowered to SYS-scope flat).
// av = LDS byte address for sub-tile 0; sub strides are 16*LDSP*4 = 4224 B.
// ---------------------------------------------------------------------------
__device__ __forceinline__ void lds_frag4(int av, v2f& b0, v2f& b1, v2f& b2, v2f& b3)
{
    unsigned long long r0, r1, r2, r3;
    asm volatile(
        "ds_load_b64 %0, %4\n\t"
        "ds_load_b64 %1, %4 offset:4224\n\t"
        "ds_load_b64 %2, %4 offset:8448\n\t"
        "ds_load_b64 %3, %4 offset:12672\n\t"
        "s_wait_dscnt 0x0"
        : "=&v"(r0), "=&v"(r1), "=&v"(r2), "=&v"(r3)
        : "v"(av));
    b0 = __builtin_bit_cast(v2f, r0);
    b1 = __builtin_bit_cast(v2f, r1);
    b2 = __builtin_bit_cast(v2f, r2);
    b3 = __builtin_bit_cast(v2f, r3);
}

// ---------------------------------------------------------------------------
// Kernel A: q[2048,4096] = x[2048,1024] @ Wq^T + bq
// 128-thread block (4 waves) computes a 64x64 tile; per 64-wide K chunk the
// TDM stages the shared 64x64 B tile (B[k][n] = Wq[n][k]) into dynamic LDS,
// then each wave runs 16 V_WMMA_F32_16X16X4_F32 steps per sub-tile.
// A frag (16x4 f32): lanes 0-15 K={k,k+1}, lanes 16-31 K={k+2,k+3} (ISA 7.12.2).
// ---------------------------------------------------------------------------
__global__ __launch_bounds__(128) void pkm_qproj(const float* __restrict__ x,
                                                 const float* __restrict__ Wq,
                                                 const float* __restrict__ bq,
                                                 float* __restrict__ q)
{
    const int lane = threadIdx.x & 31;
    const int wave = threadIdx.x >> 5;
    const int mrow = lane & 15;
    const int koff = lane >> 4;
    const int m0 = blockIdx.y * 64 + wave * 16;
    const int n0 = blockIdx.x * 64;
    const int bbase = (mrow * LDSP + (koff << 1)) * 4;   // B frag byte addr, sub 0, kk 0

    v8f acc[4] = {};
    const float* xrow = x + (size_t)(m0 + mrow) * DIMX;

    for (int k0 = 0; k0 < DIMX; k0 += KC) {
        __syncthreads();                                   // prev chunk consumed
        if (threadIdx.x < 32) {                            // wave 0 drives the TDM
            tdm_load_tile_f32(Wq + (size_t)n0 * DIMX + k0, DIMX, QCOLS, DIMX);
            __builtin_amdgcn_s_wait_tensorcnt(0);
            __asm__ __volatile__("" ::: "memory");         // DMA-complete fence
        }
        __syncthreads();                                   // tile visible

        for (int kk = 0; kk < KC; kk += 4) {
            v2f a = *(const v2f*)(xrow + k0 + kk + (koff << 1));
            v2f b0, b1, b2, b3;
            lds_frag4(bbase + kk * 4, b0, b1, b2, b3);
            acc[0] = __builtin_amdgcn_wmma_f32_16x16x4_f32(false, a, false, b0, (short)0, acc[0], false, false);
            acc[1] = __builtin_amdgcn_wmma_f32_16x16x4_f32(false, a, false, b1, (short)0, acc[1], false, false);
            acc[2] = __builtin_amdgcn_wmma_f32_16x16x4_f32(false, a, false, b2, (short)0, acc[2], false, false);
            acc[3] = __builtin_amdgcn_wmma_f32_16x16x4_f32(false, a, false, b3, (short)0, acc[3], false, false);
        }
    }

    // C/D layout: VGPR v -> M = v + 8*koff, N = mrow (ISA 7.12.2)
    #pragma unroll
    for (int sub = 0; sub < 4; ++sub) {
        const int col = n0 + sub * 16 + mrow;
        const float bias = bq[col];
        #pragma unroll
        for (int v = 0; v < 8; ++v) {
            q[(size_t)(m0 + v + 8 * koff) * QCOLS + col] = acc[sub][v] + bias;
        }
    }
}

// ---------------------------------------------------------------------------
// Kernel B: per (h,t): scores[2048,512] = q_ht[2048,256] @ keys_ht^T[256,512]
// Same 4-wave 64x64 tile structure, K = 256 in four TDM-staged chunks.
// ---------------------------------------------------------------------------
__global__ __launch_bounds__(128) void pkm_scores(const float* __restrict__ q,
                                                  const float* __restrict__ keys,
                                                  float* __restrict__ scores)
{
    const int lane = threadIdx.x & 31;
    const int wave = threadIdx.x >> 5;
    const int mrow = lane & 15;
    const int koff = lane >> 4;
    const int m0 = blockIdx.y * 64 + wave * 16;
    const int n0 = blockIdx.x * 64;
    const int ht = blockIdx.z;                     // h*2 + t, 0..15
    const int bbase = (mrow * LDSP + (koff << 1)) * 4;

    const float* arow = q + (size_t)(m0 + mrow) * QCOLS + (size_t)ht * KHALF;
    const float* kb   = keys + (size_t)ht * NKEYS * KHALF;   // [512,256] row-major

    v8f acc[4] = {};
    for (int k0 = 0; k0 < KHALF; k0 += KC) {
        __syncthreads();
        if (threadIdx.x < 32) {
            tdm_load_tile_f32(kb + (size_t)n0 * KHALF + k0, KHALF, NKEYS, KHALF);
            __builtin_amdgcn_s_wait_tensorcnt(0);
            __asm__ __volatile__("" ::: "memory");
        }
        __syncthreads();

        for (int kk = 0; kk < KC; kk += 4) {
            v2f a = *(const v2f*)(arow + k0 + kk + (koff << 1));
            v2f b0, b1, b2, b3;
            lds_frag4(bbase + kk * 4, b0, b1, b2, b3);
            acc[0] = __builtin_amdgcn_wmma_f32_16x16x4_f32(false, a, false, b0, (short)0, acc[0], false, false);
            acc[1] = __builtin_amdgcn_wmma_f32_16x16x4_f32(false, a, false, b1, (short)0, acc[1], false, false);
            acc[2] = __builtin_amdgcn_wmma_f32_16x16x4_f32(false, a, false, b2, (short)0, acc[2], false, false);
            acc[3] = __builtin_amdgcn_wmma_f32_16x16x4_f32(false, a, false, b3, (short)0, acc[3], false, false);
        }
    }

    // scores layout: [n][ht][512] ; per-n stride 16*512 = 8192
    #pragma unroll
    for (int sub = 0; sub < 4; ++sub) {
        const int col = n0 + sub * 16 + mrow;
        #pragma unroll
        for (int v = 0; v < 8; ++v) {
            scores[(size_t)(m0 + v + 8 * koff) * (16 * NKEYS)
                   + (size_t)ht * NKEYS + col] = acc[sub][v];
        }
    }
}

// ---------------------------------------------------------------------------
// Kernel C: per (n,h): top-16 of each 512-score half, 16x16 product top-16,
// softmax. Register-resident sorted lists, fully unrolled stable insert
// (strict '>' reproduces jax.lax.top_k lowest-index-first tie break).
// ---------------------------------------------------------------------------
__device__ __forceinline__ void top16_512(const float* __restrict__ s,
                                          float (&v)[KNN], int (&ix)[KNN])
{
    #pragma unroll
    for (int j = 0; j < KNN; ++j) { v[j] = -3.402823466e38f; ix[j] = 0; }
    for (int k = 0; k < NKEYS; ++k) {
        const float sv = s[k];
        if (sv > v[KNN - 1]) {
            v[KNN - 1] = sv; ix[KNN - 1] = k;
            #pragma unroll
            for (int j = KNN - 1; j > 0; --j) {
                if (v[j] > v[j - 1]) {
                    float tv = v[j]; v[j] = v[j - 1]; v[j - 1] = tv;
                    int   ti = ix[j]; ix[j] = ix[j - 1]; ix[j - 1] = ti;
                }
            }
        }
    }
}

__global__ __launch_bounds__(128) void pkm_topk(const float* __restrict__ scores,
                                                float* __restrict__ wout,
                                                int* __restrict__ iout)
{
    __shared__ float lv[128 * 32];
    __shared__ int   li[128 * 32];

    const int tid = blockIdx.x * 128 + threadIdx.x;   // 0..16383 == (n, h)
    const int n = tid >> 3;
    const int h = tid & 7;
    const float* s1 = scores + (size_t)n * (16 * NKEYS) + (size_t)(2 * h) * NKEYS;
    const float* s2 = s1 + NKEYS;

    float v1[KNN]; int i1[KNN];
    float v2[KNN]; int i2[KNN];
    top16_512(s1, v1, i1);
    top16_512(s2, v2, i2);

    // spill to LDS so the product pass can index dynamically
    float* mv = lv + threadIdx.x * 32;
    int*   mi = li + threadIdx.x * 32;
    #pragma unroll
    for (int j = 0; j < KNN; ++j) {
        mv[j] = v1[j];       mi[j] = i1[j];
        mv[j + KNN] = v2[j]; mi[j + KNN] = i2[j];
    }

    // product set visited in reference position order (a*16+b) -> stable insert matches
    float pv[KNN]; int pi[KNN];
    #pragma unroll
    for (int j = 0; j < KNN; ++j) { pv[j] = -3.402823466e38f; pi[j] = 0; }
    for (int a = 0; a < KNN; ++a) {
        const float va = mv[a];
        const int   ia = mi[a] * NKEYS;
        for (int b = 0; b < KNN; ++b) {
            const float sv = va + mv[KNN + b];
            if (sv > pv[KNN - 1]) {
                pv[KNN - 1] = sv; pi[KNN - 1] = ia + mi[KNN + b];
                #pragma unroll
                for (int j = KNN - 1; j > 0; --j) {
                    if (pv[j] > pv[j - 1]) {
                        float tv = pv[j]; pv[j] = pv[j - 1]; pv[j - 1] = tv;
                        int   ti = pi[j]; pi[j] = pi[j - 1]; pi[j - 1] = ti;
                    }
                }
            }
        }
    }

    // softmax over the 16 (sorted desc; pv[0] is the max)
    const float mx = pv[0];
    float e[KNN];
    float sum = 0.0f;
    #pragma unroll
    for (int j = 0; j < KNN; ++j) { e[j] = __expf(pv[j] - mx); sum += e[j]; }
    const float inv = 1.0f / sum;
    float* wo = wout + (size_t)tid * KNN;
    int*   io = iout + (size_t)tid * KNN;
    #pragma unroll
    for (int j = 0; j < KNN; ++j) { wo[j] = e[j] * inv; io[j] = pi[j]; }
}

// ---------------------------------------------------------------------------
// Kernel D: out[n,:] = sum_k w[n,k] * values[idx[n,k], :]   (HBM-bound stage)
// One 256-thread WG per token; float4 row loads (rows are 4KB contiguous);
// global_prefetch_b8 two rows ahead hides the dependent-index latency.
// ---------------------------------------------------------------------------
__global__ __launch_bounds__(256) void pkm_gather(const float* __restrict__ values,
                                                  const float* __restrict__ wbuf,
                                                  const int* __restrict__ ibuf,
                                                  float* __restrict__ out)
{
    __shared__ float wL[HEADS * KNN];
    __shared__ int   iL[HEADS * KNN];

    const int n = blockIdx.x;
    const int t = threadIdx.x;
    if (t < HEADS * KNN) {
        wL[t] = wbuf[(size_t)n * (HEADS * KNN) + t];
        iL[t] = ibuf[(size_t)n * (HEADS * KNN) + t];
    }
    __syncthreads();

    const float4* vbase = (const float4*)values;   // 256 float4 per row
    float4 acc = make_float4(0.f, 0.f, 0.f, 0.f);

    for (int k = 0; k < HEADS * KNN; ++k) {
        if (k + 2 < HEADS * KNN) {
            const float4* nrow = vbase + (size_t)iL[k + 2] * 256 + t;
            __builtin_prefetch((const void*)nrow, 0, 1);
        }
        const float4 v = vbase[(size_t)iL[k] * 256 + t];
        const float wgt = wL[k];
        acc.x += wgt * v.x; acc.y += wgt * v.y;
        acc.z += wgt * v.z; acc.w += wgt * v.w;
    }

    ((float4*)out)[(size_t)n * 256 + t] = acc;
}

// ---------------------------------------------------------------------------
extern "C" void kernel_launch(void* const* d_in, const int* in_sizes, int n_in,
                              void* d_out, int out_size, void* d_ws, size_t ws_size,
                              hipStream_t stream)
{
    const float* x      = (const float*)d_in[0];   // [2,1024,1024]
    const float* Wq     = (const float*)d_in[1];   // [4096,1024]
    const float* bq     = (const float*)d_in[2];   // [4096]
    const float* keys   = (const float*)d_in[3];   // [8,2,512,256]
    const float* values = (const float*)d_in[4];   // [262144,1024]
    float* out = (float*)d_out;                    // [2048,1024]

    char* ws = (char*)d_ws;
    float* q      = (float*)(ws);                         // 32 MiB
    float* scores = (float*)(ws + (32ull << 20));         // 64 MiB
    float* wbuf   = (float*)(ws + (96ull << 20));         // 1 MiB
    int*   ibuf   = (int*)  (ws + (97ull << 20));         // 1 MiB

    // Dynamic LDS (16896 B) reserves the TDM landing zone at offset 0 in A/B.
    dim3 gA(QCOLS / 64, NTOK / 64, 1);          // 64 x 32, 4-wave WGs
    pkm_qproj<<<gA, 128, TILE_LDS_BYTES, stream>>>(x, Wq, bq, q);

    dim3 gB(NKEYS / 64, NTOK / 64, 16);         // 8 x 32 x 16
    pkm_scores<<<gB, 128, TILE_LDS_BYTES, stream>>>(q, keys, scores);

    pkm_topk<<<(NTOK * HEADS) / 128, 128, 0, stream>>>(scores, wbuf, ibuf);

    pkm_gather<<<NTOK, 256, 0, stream>>>(values, wbuf, ibuf, out);
}